// AttentionSumReader_37065567764855
// MI455X (gfx1250) — compile-verified
//
#include <hip/hip_runtime.h>

typedef __attribute__((ext_vector_type(2))) float v2f;
typedef __attribute__((ext_vector_type(8))) float v8f;

#define B_ 64
#define S_ 4096
#define E_ 128
#define NUM_ENT 600
#define TILES_PER_WAVE 4
#define WAVES_PER_WG 8
#define ROWS_PER_WG (TILES_PER_WAVE * WAVES_PER_WG * 16) /* 512 */

// ---------------------------------------------------------------------------
// Kernel 1: scores[b,s] = dot(doc[b,s,:], query[b,:]) via V_WMMA_F32_16X16X4_F32
// A tile = 16 doc rows x 4 K (fp32), B = query chunk replicated in all 16 cols.
// Column 0 of the 16x16 accumulator carries the 16 dot products.
// ---------------------------------------------------------------------------
__global__ __launch_bounds__(256) void scores_wmma_kernel(
    const float* __restrict__ doc, const float* __restrict__ query,
    float* __restrict__ scores) {
  const int b = blockIdx.y;

  __shared__ float q[E_];
  if (threadIdx.x < E_) q[threadIdx.x] = query[b * E_ + threadIdx.x];
  __syncthreads();

  const int wave = threadIdx.x >> 5;
  const int lane = threadIdx.x & 31;
  const int mrow = lane & 15;        // A-matrix row this lane carries
  const int kh   = (lane >> 4) << 1; // K sub-offset: 0 (lanes 0-15), 2 (lanes 16-31)

  const int wave_row0 = blockIdx.x * ROWS_PER_WG + wave * (TILES_PER_WAVE * 16);

  for (int t = 0; t < TILES_PER_WAVE; ++t) {
    const int r0 = wave_row0 + t * 16;
    const float* arow = doc + ((size_t)b * S_ + (size_t)(r0 + mrow)) * E_ + kh;

    v8f c = {0.f, 0.f, 0.f, 0.f, 0.f, 0.f, 0.f, 0.f};
#pragma unroll
    for (int k0 = 0; k0 < E_; k0 += 4) {
      // ISA 16x4 fp32 A layout: VGPR0=K(kh), VGPR1=K(kh+1) -> contiguous 8B load
      v2f a  = *(const v2f*)(arow + k0);
      // ISA 4x16 fp32 B layout: VGPR0=row K(kh), VGPR1=row K(kh+1); all cols equal
      v2f bb = *(const v2f*)(&q[k0 + kh]);
      c = __builtin_amdgcn_wmma_f32_16x16x4_f32(
          /*neg_a=*/false, a, /*neg_b=*/false, bb,
          /*c_mod=*/(short)0, c, /*reuse_a=*/false, /*reuse_b=*/false);
    }

    // D layout: lane 0 col N=0 holds M=0..7 in v[0..7]; lane 16 holds M=8..15.
    if (mrow == 0) {
      float* outp = scores + (size_t)b * S_ + r0 + (lane >> 4) * 8;
#pragma unroll
      for (int i = 0; i < 8; ++i) outp[i] = c[i];
    }
  }
}

// ---------------------------------------------------------------------------
// Kernel 2: per batch -- masked max, exp-sum into 600 LDS entity bins + total,
// then logits[e] = log(bin[e]/total + 1e-12). One 256-thread WG per batch.
// ---------------------------------------------------------------------------
__global__ __launch_bounds__(256) void softmax_seg_log_kernel(
    const float* __restrict__ scores, const int* __restrict__ ids,
    const int* __restrict__ seqlen, float* __restrict__ out) {
  const int b = blockIdx.x;
  const int tid = threadIdx.x;
  const int len = seqlen[b];

  __shared__ float bins[NUM_ENT];
  __shared__ float red[256];

  for (int e = tid; e < NUM_ENT; e += 256) bins[e] = 0.f;

  // masked max
  float m = -3.0e38f;
  for (int s = tid; s < S_; s += 256)
    if (s < len) m = fmaxf(m, scores[(size_t)b * S_ + s]);
  red[tid] = m;
  __syncthreads();
  for (int off = 128; off > 0; off >>= 1) {
    if (tid < off) red[tid] = fmaxf(red[tid], red[tid + off]);
    __syncthreads();
  }
  const float mx = red[0];
  __syncthreads();

  // exp-sum + entity segment sum (ds_add_f32 atomics in LDS)
  float tot = 0.f;
  for (int s = tid; s < S_; s += 256) {
    if (s < len) {
      const float e = expf(scores[(size_t)b * S_ + s] - mx);
      tot += e;
      atomicAdd(&bins[ids[(size_t)b * S_ + s]], e);
    }
  }
  red[tid] = tot;
  __syncthreads();
  for (int off = 128; off > 0; off >>= 1) {
    if (tid < off) red[tid] += red[tid + off];
    __syncthreads();
  }
  const float inv = 1.0f / red[0];
  __syncthreads();

  for (int e = tid; e < NUM_ENT; e += 256)
    out[(size_t)b * NUM_ENT + e] = logf(bins[e] * inv + 1e-12f);
}

// ---------------------------------------------------------------------------
extern "C" void kernel_launch(void* const* d_in, const int* in_sizes, int n_in,
                              void* d_out, int out_size, void* d_ws, size_t ws_size,
                              hipStream_t stream) {
  const float* doc    = (const float*)d_in[0]; // [B,S,E] fp32
  const float* query  = (const float*)d_in[1]; // [B,E]   fp32
  const int*   ids    = (const int*)d_in[2];   // [B,S]   int32
  const int*   seqlen = (const int*)d_in[3];   // [B]     int32
  float* out    = (float*)d_out;               // [B,NUM_ENT] fp32
  float* scores = (float*)d_ws;                // [B,S] fp32 scratch (1 MB)

  dim3 g1(S_ / ROWS_PER_WG, B_); // 8 x 64 = 512 workgroups
  scores_wmma_kernel<<<g1, 256, 0, stream>>>(doc, query, scores);
  softmax_seg_log_kernel<<<B_, 256, 0, stream>>>(scores, ids, seqlen, out);
}